// QOC_3264175145559
// MI455X (gfx1250) — compile-verified
//
#include <hip/hip_runtime.h>

// QOC propagator for MI455X (gfx1250, wave32, WMMA f32 16x16x4 + async LDS copies).
//
// Layouts ("K-pair interleaved" so one WMMA fragment = one 8-byte LDS load):
//   A-layout (left operand):  element (row,k) at ((k>>1)*64 + row)*2 + (k&1)
//   B-layout (right operand): element (k,col) at ((k>>1)*64 + col)*2 + (k&1)
// In the product tree, the matrix produced at pair index p is the LEFT operand
// of the next level iff p is even -> every producer stores in its consumer's
// layout, so the chain kernel's LDS fill is a straight contiguous async copy.

typedef __attribute__((ext_vector_type(2))) float v2f;
typedef __attribute__((ext_vector_type(8))) float v8f;

#define QDT     0.001f
#define DD      64
#define MSZ     (DD * DD)      // 4096 floats per plane
#define NSTEPS  4096
#define NCTRL   8

#define PKA(row, col) (((((col) >> 1) * DD) + (row)) * 2 + ((col) & 1))  // left/A
#define PKB(row, col) (((((row) >> 1) * DD) + (col)) * 2 + ((row) & 1))  // right/B

static __device__ __forceinline__ v8f wmma4(v2f a, v2f b, v8f c) {
    return __builtin_amdgcn_wmma_f32_16x16x4_f32(
        /*neg_a=*/false, a, /*neg_b=*/false, b,
        /*c_mod=*/(short)0, c, /*reuse_a=*/false, /*reuse_b=*/false);
}

// Fragment load: K-values {k0+2*half, k0+2*half+1} for non-K index (j16+l).
// One contiguous 8-byte LDS load -> even-aligned VGPR pair, no shuffles.
static __device__ __forceinline__ v2f ldF(const float* P, int j16, int k0,
                                          int half, int l) {
    return *(const v2f*)&P[((k0 >> 1) + half) * (2 * DD) + 2 * (j16 + l)];
}

// One 64x64 complex matmul: this wave's two 16x16 complex C tiles.
static __device__ __forceinline__ void cmatmul_tiles(
    const float* XAre, const float* XAim,
    const float* TBre, const float* TBim,
    int tm16, int tn0, int half, int lr,
    v8f& cr0, v8f& ci0, v8f& cr1, v8f& ci1)
{
    #pragma unroll
    for (int k0 = 0; k0 < DD; k0 += 4) {
        v2f Ar = ldF(XAre, tm16, k0, half, lr);
        v2f Ai = ldF(XAim, tm16, k0, half, lr);
        v2f AiN; AiN.x = -Ai.x; AiN.y = -Ai.y;   // f32 WMMA: only C-neg exists
        v2f Br0 = ldF(TBre, tn0,      k0, half, lr);
        v2f Bi0 = ldF(TBim, tn0,      k0, half, lr);
        v2f Br1 = ldF(TBre, tn0 + 16, k0, half, lr);
        v2f Bi1 = ldF(TBim, tn0 + 16, k0, half, lr);
        cr0 = wmma4(Ar,  Br0, cr0);  cr0 = wmma4(AiN, Bi0, cr0);
        ci0 = wmma4(Ar,  Bi0, ci0);  ci0 = wmma4(Ai,  Br0, ci0);
        cr1 = wmma4(Ar,  Br1, cr1);  cr1 = wmma4(AiN, Bi1, cr1);
        ci1 = wmma4(Ar,  Bi1, ci1);  ci1 = wmma4(Ai,  Br1, ci1);
    }
}

// ---------------------------------------------------------------------------
// Kernel 1: U[n] = expm(-i*H[n]*DT), order-6 Taylor via Horner (5 matmuls).
// One 256-thread block (8 waves) per timestep n.
// ---------------------------------------------------------------------------
__global__ __launch_bounds__(256)
void qoc_expm(const float* __restrict__ Hd_re, const float* __restrict__ Hd_im,
              const float* __restrict__ Hc_re, const float* __restrict__ Hc_im,
              const float* __restrict__ actl,  float* __restrict__ U)
{
    __shared__ __align__(16) float XAre[MSZ];   // X, A-layout
    __shared__ __align__(16) float XAim[MSZ];
    __shared__ __align__(16) float TBre[MSZ];   // Horner accumulator, B-layout
    __shared__ __align__(16) float TBim[MSZ];

    const int n   = blockIdx.x;
    const int tid = threadIdx.x;

    // --- X = -i*H[n]*DT ;  T = I + X/6 ---------------------------------------
    // X_re =  DT*(Hd_im + sum_l a_l*Hc_im) ; X_im = -DT*(Hd_re + sum_l a_l*Hc_re)
    float av[NCTRL];
    #pragma unroll
    for (int l = 0; l < NCTRL; ++l) av[l] = actl[l * NSTEPS + n];  // uniform

    #pragma unroll
    for (int j = 0; j < 16; j += 2) {
        const int e = tid * 16 + j;
        const int r = e >> 6, c = e & 63;       // c is even
        float sr0 = Hd_im[e],     si0 = Hd_re[e];
        float sr1 = Hd_im[e + 1], si1 = Hd_re[e + 1];
        #pragma unroll
        for (int l = 0; l < NCTRL; ++l) {
            sr0 = fmaf(av[l], Hc_im[l * MSZ + e],     sr0);
            si0 = fmaf(av[l], Hc_re[l * MSZ + e],     si0);
            sr1 = fmaf(av[l], Hc_im[l * MSZ + e + 1], sr1);
            si1 = fmaf(av[l], Hc_re[l * MSZ + e + 1], si1);
        }
        v2f xr; xr.x =  QDT * sr0; xr.y =  QDT * sr1;
        v2f xi; xi.x = -QDT * si0; xi.y = -QDT * si1;
        *(v2f*)&XAre[PKA(r, c)] = xr;           // (r,c),(r,c+1) contiguous in A
        *(v2f*)&XAim[PKA(r, c)] = xi;
        TBre[PKB(r, c)]     = xr.x * (1.0f / 6.0f) + ((r == c)     ? 1.0f : 0.0f);
        TBre[PKB(r, c + 1)] = xr.y * (1.0f / 6.0f) + ((r == c + 1) ? 1.0f : 0.0f);
        TBim[PKB(r, c)]     = xi.x * (1.0f / 6.0f);
        TBim[PKB(r, c + 1)] = xi.y * (1.0f / 6.0f);
    }
    __syncthreads();

    const int lane = tid & 31;
    const int wave = tid >> 5;
    const int half = lane >> 4;
    const int lr   = lane & 15;
    const int tm16 = (wave >> 1) * 16;
    const int tn0  = (wave & 1) * 32;

    // --- 5 Horner steps: T <- I + (1/m) * X @ T, m = 5..1 --------------------
    for (int step = 0; step < 5; ++step) {
        const float s = 1.0f / (float)(5 - step);
        v8f cr0 = {}, ci0 = {}, cr1 = {}, ci1 = {};
        cmatmul_tiles(XAre, XAim, TBre, TBim, tm16, tn0, half, lr,
                      cr0, ci0, cr1, ci1);
        __syncthreads();                        // all reads of T done

        if (step < 4) {
            #pragma unroll
            for (int i = 0; i < 8; i += 2) {    // rows i,i+1 -> one b64 store
                const int row = tm16 + 8 * half + i;     // even
                const int c0  = tn0 + lr;
                const int c1  = tn0 + 16 + lr;
                v2f w;
                w.x = s * cr0[i]     + ((row == c0)     ? 1.0f : 0.0f);
                w.y = s * cr0[i + 1] + ((row + 1 == c0) ? 1.0f : 0.0f);
                *(v2f*)&TBre[PKB(row, c0)] = w;
                w.x = s * ci0[i];     w.y = s * ci0[i + 1];
                *(v2f*)&TBim[PKB(row, c0)] = w;
                w.x = s * cr1[i]     + ((row == c1)     ? 1.0f : 0.0f);
                w.y = s * cr1[i + 1] + ((row + 1 == c1) ? 1.0f : 0.0f);
                *(v2f*)&TBre[PKB(row, c1)] = w;
                w.x = s * ci1[i];     w.y = s * ci1[i + 1];
                *(v2f*)&TBim[PKB(row, c1)] = w;
            }
            __syncthreads();
        } else {
            // U[n] = I + X@T4 -> global, in the layout level-0 will consume
            float* gr = U + (size_t)n * (2 * MSZ);
            float* gi = gr + MSZ;
            const bool left = (n & 1) == 0;
            #pragma unroll
            for (int i = 0; i < 8; ++i) {
                const int row = tm16 + 8 * half + i;
                const int c0  = tn0 + lr;
                const int c1  = tn0 + 16 + lr;
                const int i0  = left ? PKA(row, c0) : PKB(row, c0);
                const int i1  = left ? PKA(row, c1) : PKB(row, c1);
                gr[i0] = cr0[i] + ((row == c0) ? 1.0f : 0.0f);
                gi[i0] = ci0[i];
                gr[i1] = cr1[i] + ((row == c1) ? 1.0f : 0.0f);
                gi[i1] = ci1[i];
            }
        }
    }
}

// ---------------------------------------------------------------------------
// Kernel 2: one tree level. Block p: P[2p*s] <- P[2p*s] @ P[(2p+1)*s].
// LDS fill via CDNA5 async global->LDS copies (ASYNCcnt), contiguous planes.
// ---------------------------------------------------------------------------
__global__ __launch_bounds__(256)
void qoc_chain(float* __restrict__ U, int stride)
{
    __shared__ __align__(16) float XAre[MSZ];   // left matrix (A-layout)
    __shared__ __align__(16) float XAim[MSZ];
    __shared__ __align__(16) float TBre[MSZ];   // right matrix (B-layout)
    __shared__ __align__(16) float TBim[MSZ];

    const int p   = blockIdx.x;
    const int tid = threadIdx.x;
    float*       gA = U + (size_t)(2 * p)     * stride * (2 * MSZ);
    const float* gB = U + (size_t)(2 * p + 1) * stride * (2 * MSZ);

    // ---- async fill: 4 planes x 16 KB, 64 B (4 x b128) per thread per plane
    {
        const unsigned off = (unsigned)tid * 64u;   // byte offset within plane
        const unsigned dar = (unsigned)(unsigned long long)&XAre[0] + off;
        const unsigned dai = (unsigned)(unsigned long long)&XAim[0] + off;
        const unsigned dbr = (unsigned)(unsigned long long)&TBre[0] + off;
        const unsigned dbi = (unsigned)(unsigned long long)&TBim[0] + off;
        const float* sar = gA;
        const float* sai = gA + MSZ;
        const float* sbr = gB;
        const float* sbi = gB + MSZ;
        #pragma unroll
        for (int q = 0; q < 4; ++q) {
            const unsigned qb = (unsigned)q * 16u;
            asm volatile("global_load_async_to_lds_b128 %0, %1, %2"
                         :: "v"(dar + qb), "v"(off + qb), "s"(sar) : "memory");
            asm volatile("global_load_async_to_lds_b128 %0, %1, %2"
                         :: "v"(dai + qb), "v"(off + qb), "s"(sai) : "memory");
            asm volatile("global_load_async_to_lds_b128 %0, %1, %2"
                         :: "v"(dbr + qb), "v"(off + qb), "s"(sbr) : "memory");
            asm volatile("global_load_async_to_lds_b128 %0, %1, %2"
                         :: "v"(dbi + qb), "v"(off + qb), "s"(sbi) : "memory");
        }
        asm volatile("s_wait_asynccnt 0" ::: "memory");
    }
    __syncthreads();

    const int lane = tid & 31;
    const int wave = tid >> 5;
    const int half = lane >> 4;
    const int lr   = lane & 15;
    const int tm16 = (wave >> 1) * 16;
    const int tn0  = (wave & 1) * 32;

    v8f cr0 = {}, ci0 = {}, cr1 = {}, ci1 = {};
    cmatmul_tiles(XAre, XAim, TBre, TBim, tm16, tn0, half, lr,
                  cr0, ci0, cr1, ci1);
    __syncthreads();   // gA fully resident in LDS before overwrite

    // write result into gA's slot, in the layout the NEXT level consumes
    float* gr = gA;
    float* gi = gA + MSZ;
    const bool left = (p & 1) == 0;
    #pragma unroll
    for (int i = 0; i < 8; ++i) {
        const int row = tm16 + 8 * half + i;
        const int c0  = tn0 + lr;
        const int c1  = tn0 + 16 + lr;
        const int i0  = left ? PKA(row, c0) : PKB(row, c0);
        const int i1  = left ? PKA(row, c1) : PKB(row, c1);
        gr[i0] = cr0[i];
        gi[i0] = ci0[i];
        gr[i1] = cr1[i];
        gi[i1] = ci1[i];
    }
}

// ---------------------------------------------------------------------------
// Kernel 3: final matrix (A-layout, slot 0) -> interleaved complex64 in d_out.
// ---------------------------------------------------------------------------
__global__ __launch_bounds__(256)
void qoc_interleave(const float* __restrict__ U, float* __restrict__ out)
{
    const int e = blockIdx.x * 256 + threadIdx.x;   // 0..4095
    const int r = e >> 6, c = e & 63;
    const int idx = PKA(r, c);
    out[2 * e + 0] = U[idx];
    out[2 * e + 1] = U[MSZ + idx];
}

extern "C" void kernel_launch(void* const* d_in, const int* in_sizes, int n_in,
                              void* d_out, int out_size, void* d_ws, size_t ws_size,
                              hipStream_t stream)
{
    (void)in_sizes; (void)n_in; (void)out_size; (void)ws_size;

    const float* Hd_re = (const float*)d_in[0];   // [64,64]
    const float* Hd_im = (const float*)d_in[1];   // [64,64]
    const float* Hc_re = (const float*)d_in[2];   // [8,64,64]
    const float* Hc_im = (const float*)d_in[3];   // [8,64,64]
    const float* actl  = (const float*)d_in[4];   // [8,4096]

    float* U = (float*)d_ws;   // 4096 * 32 KiB = 128 MiB (L2-resident on MI455X)

    qoc_expm<<<NSTEPS, 256, 0, stream>>>(Hd_re, Hd_im, Hc_re, Hc_im, actl, U);

    for (int stride = 1; stride < NSTEPS; stride <<= 1) {
        const int pairs = NSTEPS / (2 * stride);
        qoc_chain<<<pairs, 256, 0, stream>>>(U, stride);
    }

    qoc_interleave<<<MSZ / 256, 256, 0, stream>>>(U, (float*)d_out);
}